// LQRClass_89421219103569
// MI455X (gfx1250) — compile-verified
//
#include <hip/hip_runtime.h>

// LQR KKT solve via Riccati recursion on gfx1250 (MI455X).
// One wave32 workgroup; per-step 16x16 f32 matmuls via V_WMMA_F32_16X16X4_F32.
// C_t tiles are double-buffered in LDS via GLOBAL_LOAD_ASYNC_TO_LDS_B128
// (ASYNCcnt) so HBM latency overlaps the serial Riccati chain.
// d_out = [ z (128*24) | mu (128*16) ] = 5120 floats.

typedef __attribute__((ext_vector_type(2))) float v2f;
typedef __attribute__((ext_vector_type(8))) float v8f;
typedef __attribute__((ext_vector_type(4))) int v4i;

#define T_HOR 128
#define NS 16   // n_state
#define NC 8    // n_ctrl
#define NA 24   // n_all
#define NP 32   // n_all padded to 2x16 tiles

#if defined(__has_builtin)
#if __has_builtin(__builtin_amdgcn_global_load_async_to_lds_b128)
#define USE_ASYNC_LDS 1
#endif
#endif

#ifdef USE_ASYNC_LDS
// Builtin signature (from clang diagnostic): param0 = v4i addrspace(1)* (global),
// param1 = v4i addrspace(3)* (LDS), then imm offset, imm cpol.
typedef __attribute__((address_space(1))) v4i as1_v4i;
typedef __attribute__((address_space(3))) v4i as3_v4i;
#endif

__device__ __forceinline__ void wait_async0() {
#ifdef USE_ASYNC_LDS
#if __has_builtin(__builtin_amdgcn_s_wait_asynccnt)
  __builtin_amdgcn_s_wait_asynccnt(0);
#else
  asm volatile("s_wait_asynccnt 0x0" ::: "memory");
#endif
#endif
}

// Issue async copy of one 24x24 C_t tile (row-major, 96B rows) into the
// interior of a zero-padded 32-stride LDS buffer. 144 x 16B chunks, 32 lanes.
__device__ __forceinline__ void issue_ct_load(const float* Ct, float* Qbuf, int lane) {
#ifdef USE_ASYNC_LDS
#pragma unroll
  for (int rnd = 0; rnd < 5; ++rnd) {
    int m = rnd * 32 + lane;
    if (m < 144) {
      int r = m / 6, j = m % 6;
      __builtin_amdgcn_global_load_async_to_lds_b128(
          (as1_v4i*)(Ct + r * NA + j * 4), (as3_v4i*)(Qbuf + r * NP + j * 4), 0, 0);
    }
  }
#endif
}

// ---- WMMA f32 16x16x4 fragment helpers (wave32) -------------------------
// A (16x4 f32): lanes 0-15 hold rows M=0..15 with {K=0,K=1}; lanes 16-31 hold {K=2,K=3}.
__device__ __forceinline__ v2f load_a_frag(const float* X, int ld, int r0, int k0) {
  int lane = threadIdx.x & 31;
  int half = lane >> 4, r = (lane & 15) + r0;
  int k = k0 + 2 * half;
  v2f a; a.x = X[r * ld + k]; a.y = X[r * ld + k + 1];
  return a;
}
// B (4x16 f32): lane = column; lanes 0-15 hold {K=0,K=1}; lanes 16-31 hold {K=2,K=3}.
__device__ __forceinline__ v2f load_b_frag(const float* Y, int ld, int k0, int c0) {
  int lane = threadIdx.x & 31;
  int half = lane >> 4, c = (lane & 15) + c0;
  int k = k0 + 2 * half;
  v2f b; b.x = Y[k * ld + c]; b.y = Y[(k + 1) * ld + c];
  return b;
}
// C/D (16x16 f32): VGPR r holds row r (lanes 0-15) / row r+8 (lanes 16-31).
__device__ __forceinline__ v8f load_c_frag(const float* X, int ld, int r0, int c0) {
  int lane = threadIdx.x & 31;
  int half = lane >> 4, c = (lane & 15) + c0;
  v8f acc;
#pragma unroll
  for (int r = 0; r < 8; ++r) acc[r] = X[(r0 + r + 8 * half) * ld + c];
  return acc;
}
__device__ __forceinline__ void store_d_frag(float* X, int ld, int r0, int c0, v8f d) {
  int lane = threadIdx.x & 31;
  int half = lane >> 4, c = (lane & 15) + c0;
#pragma unroll
  for (int r = 0; r < 8; ++r) X[(r0 + r + 8 * half) * ld + c] = d[r];
}

// acc += X_tile(16x16) * Y_tile(16x16), K=16 via 4 chained K=4 WMMAs.
__device__ __forceinline__ v8f mm16(const float* X, int ldx, int xr, int xc,
                                    const float* Y, int ldy, int yr, int yc, v8f acc) {
#pragma unroll
  for (int k0 = 0; k0 < 16; k0 += 4) {
    v2f a = load_a_frag(X, ldx, xr, xc + k0);
    v2f b = load_b_frag(Y, ldy, yr + k0, yc);
    acc = __builtin_amdgcn_wmma_f32_16x16x4_f32(false, a, false, b, (short)0, acc,
                                                false, false);
  }
  return acc;
}
// Same with pre-hoisted (loop-invariant) B fragments.
__device__ __forceinline__ v8f mm16_bp(const float* X, int ldx, int xr, int xc,
                                       const v2f* bf, v8f acc) {
#pragma unroll
  for (int k = 0; k < 4; ++k) {
    v2f a = load_a_frag(X, ldx, xr, xc + 4 * k);
    acc = __builtin_amdgcn_wmma_f32_16x16x4_f32(false, a, false, bf[k], (short)0, acc,
                                                false, false);
  }
  return acc;
}
// Same with pre-hoisted (loop-invariant) A fragments.
__device__ __forceinline__ v8f mm16_ap(const v2f* af, const float* Y, int ldy, int yr,
                                       int yc, v8f acc) {
#pragma unroll
  for (int k = 0; k < 4; ++k) {
    v2f b = load_b_frag(Y, ldy, yr + 4 * k, yc);
    acc = __builtin_amdgcn_wmma_f32_16x16x4_f32(false, af[k], false, b, (short)0, acc,
                                                false, false);
  }
  return acc;
}

__global__ __launch_bounds__(32) void lqr_riccati_kernel(
    const float* __restrict__ A, const float* __restrict__ B,
    const float* __restrict__ x0, const float* __restrict__ C,
    const float* __restrict__ c, float* __restrict__ out, float* __restrict__ ws) {
  __shared__ float Qb[2][NP * NP];  // double-buffered padded Q_t (32x32)
  __shared__ float F[NS * NP];      // [A | B | 0]  16x32
  __shared__ float FT[NP * NS];     // F transpose  32x16
  __shared__ float Vl[NS * NS];     // cost-to-go V_{t+1} -> V_t
  __shared__ float vl[NS];
  __shared__ float G[NS * NP];      // V * F
  __shared__ float qv[NP];
  __shared__ float Aug[NC * 16];    // Gauss-Jordan [Quu | I]
  __shared__ float Wp[NS * NS];     // padded -Quu^{-1}
  __shared__ float Km[NS * NS];     // padded gain [K ; 0]
  __shared__ float kv[NC];
  __shared__ float xs[NS];
  __shared__ float us[NC];

  const int lane = threadIdx.x;

  // --- init F = [A B 0], FT, V_T = 0, v_T = 0, zero both Q buffers ---
  for (int idx = lane; idx < NS * NP; idx += 32) {
    int r = idx / NP, cc = idx % NP;
    float val = 0.f;
    if (cc < NS) val = A[r * NS + cc];
    else if (cc < NA) val = B[r * NC + (cc - NS)];
    F[idx] = val;
    FT[cc * NS + r] = val;
  }
  for (int idx = lane; idx < 2 * NP * NP; idx += 32) Qb[0][idx] = 0.f;
  for (int idx = lane; idx < NS * NS; idx += 32) Vl[idx] = 0.f;
  if (lane < NS) vl[lane] = 0.f;
  __syncthreads();

  // Hoist loop-invariant fragments: B-frags of F, A-frags of FT.
  v2f fB[2][4], ftA[2][4];
#pragma unroll
  for (int blk = 0; blk < 2; ++blk)
#pragma unroll
    for (int k = 0; k < 4; ++k) {
      fB[blk][k] = load_b_frag(F, NP, 4 * k, 16 * blk);
      ftA[blk][k] = load_a_frag(FT, NS, 16 * blk, 4 * k);
    }

  float* wsV = ws;                     // T*256
  float* wsv = wsV + T_HOR * NS * NS;  // T*16
  float* wsK = wsv + T_HOR * NS;       // T*128
  float* wsk = wsK + T_HOR * NC * NS;  // T*8

#ifdef USE_ASYNC_LDS
  issue_ct_load(C + (size_t)(T_HOR - 1) * NA * NA, &Qb[(T_HOR - 1) & 1][0], lane);
#endif

  // ================= backward Riccati sweep =================
  for (int t = T_HOR - 1; t >= 0; --t) {
    float* Q = &Qb[t & 1][0];
#ifdef USE_ASYNC_LDS
    wait_async0();   // C_t now resident in Q's interior (padding invariantly 0)
    __syncthreads();
    if (t > 0) issue_ct_load(C + (size_t)(t - 1) * NA * NA, &Qb[(t - 1) & 1][0], lane);
#else
    const float* Ct = C + (size_t)t * NA * NA;
    for (int idx = lane; idx < NP * NP; idx += 32) {
      int r = idx / NP, cc = idx % NP;
      Q[idx] = (r < NA && cc < NA) ? Ct[r * NA + cc] : 0.f;
    }
    __syncthreads();
#endif

    // G = V_{t+1} * F   (16x32, two tiles; F frags pre-hoisted)
    {
      v8f d0 = {};
      v8f d1 = {};
      d0 = mm16_bp(Vl, NS, 0, 0, fB[0], d0);
      d1 = mm16_bp(Vl, NS, 0, 0, fB[1], d1);
      store_d_frag(G, NP, 0, 0, d0);
      store_d_frag(G, NP, 0, 16, d1);
    }
    __syncthreads();

    // Q += F^T * G   (32x32, four tiles; FT frags pre-hoisted)
#pragma unroll
    for (int bi = 0; bi < 2; ++bi) {
#pragma unroll
      for (int bj = 0; bj < 2; ++bj) {
        v8f acc = load_c_frag(Q, NP, bi * 16, bj * 16);
        acc = mm16_ap(ftA[bi], G, NP, 0, bj * 16, acc);
        store_d_frag(Q, NP, bi * 16, bj * 16, acc);
      }
    }
    __syncthreads();

    // q = pad(c_t) + F^T v_{t+1}   (32-vector, one element per lane)
    {
      float s = (lane < NA) ? c[(size_t)t * NA + lane] : 0.f;
#pragma unroll
      for (int k = 0; k < NS; ++k) s += FT[lane * NS + k] * vl[k];
      qv[lane] = s;
    }
    __syncthreads();

    // Invert Quu (8x8 SPD) via Gauss-Jordan on [Quu | I], lanes 0-15 = columns
    if (lane < 16) {
#pragma unroll
      for (int r = 0; r < NC; ++r)
        Aug[r * 16 + lane] =
            (lane < NC) ? Q[(16 + r) * NP + 16 + lane] : ((lane - NC) == r ? 1.f : 0.f);
    }
    __syncthreads();
    for (int p = 0; p < NC; ++p) {
      float pivinv = 1.f / Aug[p * 16 + p];
      __syncthreads();
      if (lane < 16) Aug[p * 16 + lane] *= pivinv;
      __syncthreads();
#pragma unroll
      for (int r = 0; r < NC; ++r) {
        if (r == p) continue;
        float f = Aug[r * 16 + p];
        __syncthreads();
        if (lane < 16) Aug[r * 16 + lane] -= f * Aug[p * 16 + lane];
        __syncthreads();
      }
    }

    // Wp = pad(-Quu^{-1})
    for (int idx = lane; idx < NS * NS; idx += 32) {
      int r = idx / NS, cc = idx % NS;
      Wp[idx] = (r < NC && cc < NC) ? -Aug[r * 16 + 8 + cc] : 0.f;
    }
    __syncthreads();

    // Km = Wp * Q[16:32, 0:16]  ->  [K_t ; 0]   (rows >=8 stay zero)
    {
      v8f d = {};
      d = mm16(Wp, NS, 0, 0, Q, NP, 16, 0, d);
      store_d_frag(Km, NS, 0, 0, d);
    }
    // k_t = -Quu^{-1} * q_u
    if (lane < NC) {
      float s = 0.f;
#pragma unroll
      for (int j = 0; j < NC; ++j) s -= Aug[lane * 16 + 8 + j] * qv[16 + j];
      kv[lane] = s;
    }
    __syncthreads();

    // V_t = Qxx + Qxu * K_t   (tile(0,1) has zero cols >=24, Km zero rows >=8)
    {
      v8f acc = load_c_frag(Q, NP, 0, 0);
      acc = mm16(Q, NP, 0, 16, Km, NS, 0, 0, acc);
      store_d_frag(Vl, NS, 0, 0, acc);
    }
    // v_t = q_x + Qxu * k_t
    if (lane < NS) {
      float s = qv[lane];
#pragma unroll
      for (int j = 0; j < NC; ++j) s += Q[lane * NP + 16 + j] * kv[j];
      vl[lane] = s;
    }
    __syncthreads();

    // stash gains & value for the forward pass
    for (int idx = lane; idx < NS * NS; idx += 32) wsV[t * NS * NS + idx] = Vl[idx];
    if (lane < NS) wsv[t * NS + lane] = vl[lane];
    for (int idx = lane; idx < NC * NS; idx += 32) wsK[t * NC * NS + idx] = Km[idx];
    if (lane < NC) wsk[t * NC + lane] = kv[lane];
    __syncthreads();
  }

  // ================= forward rollout =================
  float* z = out;                // 128*24
  float* mu = out + T_HOR * NA;  // 128*16
  if (lane < NS) xs[lane] = x0[lane];
  __syncthreads();
  for (int t = 0; t < T_HOR; ++t) {
    if (lane < NS) z[t * NA + lane] = xs[lane];
    if (lane < NC) {  // u_t = K_t x_t + k_t
      float s = wsk[t * NC + lane];
#pragma unroll
      for (int i = 0; i < NS; ++i) s += wsK[t * NC * NS + lane * NS + i] * xs[i];
      us[lane] = s;
      z[t * NA + NS + lane] = s;
    }
    __syncthreads();
    if (lane < NS) {  // mu_t = +-(V_t x_t + v_t); t=0 row of BL carries +I -> sign flip
      float s = wsv[t * NS + lane];
#pragma unroll
      for (int k = 0; k < NS; ++k) s += wsV[t * NS * NS + lane * NS + k] * xs[k];
      mu[t * NS + lane] = (t == 0) ? -s : s;
    }
    float xn = 0.f;
    if (lane < NS) {  // x_{t+1} = A x_t + B u_t  (read from F = [A B 0])
#pragma unroll
      for (int k = 0; k < NS; ++k) xn += F[lane * NP + k] * xs[k];
#pragma unroll
      for (int j = 0; j < NC; ++j) xn += F[lane * NP + NS + j] * us[j];
    }
    __syncthreads();
    if (lane < NS) xs[lane] = xn;
    __syncthreads();
  }
}

extern "C" void kernel_launch(void* const* d_in, const int* in_sizes, int n_in,
                              void* d_out, int out_size, void* d_ws, size_t ws_size,
                              hipStream_t stream) {
  const float* A = (const float*)d_in[0];   // 16x16
  const float* B = (const float*)d_in[1];   // 16x8
  const float* x0 = (const float*)d_in[2];  // 16
  const float* C = (const float*)d_in[3];   // 128x24x24
  const float* c = (const float*)d_in[4];   // 128x24
  // d_in[5] = T (hardcoded 128)
  float* out = (float*)d_out;  // 5120 floats
  float* ws = (float*)d_ws;    // needs 52224 floats (~209 KB)
  lqr_riccati_kernel<<<dim3(1), dim3(32), 0, stream>>>(A, B, x0, C, c, out, ws);
}